// HeteroLayerNorm_62156766708296
// MI455X (gfx1250) — compile-verified
//
#include <hip/hip_runtime.h>
#include <stdint.h>

// ---------------------------------------------------------------------------
// HeteroLayerNorm on MI455X (gfx1250)
//   N = 1,000,000 rows, C = 128 features (f32), T = 64 types (int64 ids)
// Memory-bound: 2 streaming passes over 512MB of x  ->  ~66us roofline.
// CDNA5 paths used: ds_add_f32 LDS atomics, non-temporal vmem hints,
// Tensor Data Mover (tensor_load_to_lds) to stage the stats table into LDS.
// ---------------------------------------------------------------------------

typedef float v4f  __attribute__((ext_vector_type(4)));
typedef unsigned int u32x4 __attribute__((ext_vector_type(4)));
typedef int  i32x8 __attribute__((ext_vector_type(8)));
typedef int  i32x4 __attribute__((ext_vector_type(4)));

#define CDIM 128
#define TDIM 64
#define EPS_ 1e-5f

// ---------------- kernel 0: zero the accumulator workspace ----------------
__global__ void hln_zero(float* __restrict__ ws, int n) {
  int i = blockIdx.x * blockDim.x + threadIdx.x;
  if (i < n) ws[i] = 0.0f;
}

// ---------------- kernel 1: per-block LDS-binned partial sums --------------
// Block = 256 threads = 8 waves. One wave owns one row per iteration:
// lane l loads float4 x[row, 4l..4l+3] (B128, non-temporal) and ds_add_f32's
// into the shared sum/sumsq bins for that row's type. Bank-conflict free.
__global__ __launch_bounds__(256) void hln_partial(
    const float* __restrict__ x, const long long* __restrict__ tv,
    float* __restrict__ g_sum, float* __restrict__ g_sq,
    float* __restrict__ g_cnt, int N) {
  extern __shared__ float smem[];            // [8192 sum | 8192 sq | 64 cnt]
  float* s_sum = smem;
  float* s_sq  = smem + TDIM * CDIM;
  float* s_cnt = smem + 2 * TDIM * CDIM;

  for (int i = threadIdx.x; i < 2 * TDIM * CDIM + TDIM; i += blockDim.x)
    smem[i] = 0.0f;
  __syncthreads();

  const int wave = threadIdx.x >> 5;
  const int lane = threadIdx.x & 31;
  const long long stride = (long long)(blockDim.x >> 5) * gridDim.x;

  for (long long row = (long long)blockIdx.x * 8 + wave; row < N; row += stride) {
    const int t = (int)tv[row];
    const v4f v = __builtin_nontemporal_load(
        (const v4f*)(x + (size_t)row * CDIM + lane * 4));
    float* bs = s_sum + t * CDIM + lane * 4;
    float* bq = s_sq  + t * CDIM + lane * 4;
    atomicAdd(bs + 0, v.x);        atomicAdd(bs + 1, v.y);
    atomicAdd(bs + 2, v.z);        atomicAdd(bs + 3, v.w);
    atomicAdd(bq + 0, v.x * v.x);  atomicAdd(bq + 1, v.y * v.y);
    atomicAdd(bq + 2, v.z * v.z);  atomicAdd(bq + 3, v.w * v.w);
    if (lane == 0) atomicAdd(s_cnt + t, 1.0f);
  }
  __syncthreads();

  // flush block partials to global (tables are tiny: stay hot in L2)
  for (int i = threadIdx.x; i < TDIM * CDIM; i += blockDim.x) {
    atomicAdd(g_sum + i, s_sum[i]);
    atomicAdd(g_sq  + i, s_sq[i]);
  }
  for (int i = threadIdx.x; i < TDIM; i += blockDim.x)
    atomicAdd(g_cnt + i, s_cnt[i]);
}

// ---------------- kernel 2: finalize mean / inv-std ------------------------
__global__ void hln_finalize(const float* __restrict__ g_sum,
                             const float* __restrict__ g_sq,
                             const float* __restrict__ g_cnt,
                             float* __restrict__ g_stats /* [mean|istd] */) {
  int i = blockIdx.x * blockDim.x + threadIdx.x;
  if (i >= TDIM * CDIM) return;
  int t = i >> 7;                                   // i / CDIM
  float cnt  = fmaxf(g_cnt[t], 1.0f);
  float mean = g_sum[i] / cnt;
  float var  = fmaxf(g_sq[i] / cnt - mean * mean, 0.0f);
  g_stats[i]               = mean;                  // mean table
  g_stats[TDIM * CDIM + i] = 1.0f / sqrtf(var + EPS_);  // inv-std table
}

// ---------------- kernel 3: gather-normalize with TDM-staged stats ---------
// Wave 0 DMAs the whole 64KB [mean|istd] table into LDS with one
// tensor_load_to_lds (1D D#: data_size=4B, tile_dim0=16384 elements,
// lds_addr=0 == base of dynamic LDS), waits TENSORcnt, then the block
// streams rows: non-temporal B128 load, LDS gather of the row-type stats,
// fused (x-mean)*istd, non-temporal B128 store.
__global__ __launch_bounds__(256) void hln_normalize(
    const float* __restrict__ x, const long long* __restrict__ tv,
    const float* __restrict__ g_stats, float* __restrict__ out, int N) {
  extern __shared__ float smem[];                   // 16384 floats
  if ((threadIdx.x >> 5) == 0) {
    const uint64_t ga = (uint64_t)(uintptr_t)g_stats;
    u32x4 g0;
    g0.x = 1u;                                      // count=1 (valid user D#)
    g0.y = 0u;                                      // lds_addr = 0 (dyn-LDS base)
    g0.z = (uint32_t)ga;                            // global_addr[31:0]
    g0.w = (uint32_t)(ga >> 32) | (2u << 30);       // global_addr[56:32] | type=2
    i32x8 g1;
    g1[0] = (int)(2u << 16);                        // wg_mask=0, data_size=4B
    g1[1] = (int)((16384u & 0xFFFFu) << 16);        // tensor_dim0[15:0]
    g1[2] = (int)((16384u >> 16) | (1u << 16));     // tensor_dim0[31:16] | tensor_dim1=1
    g1[3] = (int)(16384u << 16);                    // tensor_dim1 hi | tile_dim0=16384
    g1[4] = 1;                                      // tile_dim1=1, tile_dim2=0
    g1[5] = 16384;                                  // tensor_dim0_stride[31:0]
    g1[6] = 0;                                      // stride0 hi | stride1 lo
    g1[7] = 0;                                      // stride1 hi
    i32x4 zg = (i32x4)0;                            // groups 2/3: unused dims
    asm volatile("tensor_load_to_lds %0, %1, %2, %3"
                 :: "s"(g0), "s"(g1), "s"(zg), "s"(zg) : "memory");
    __builtin_amdgcn_s_wait_tensorcnt(0);
  }
  __syncthreads();

  const float* s_mean = smem;
  const float* s_istd = smem + TDIM * CDIM;
  const int wave = threadIdx.x >> 5;
  const int lane = threadIdx.x & 31;
  const long long stride = (long long)(blockDim.x >> 5) * gridDim.x;

  for (long long row = (long long)blockIdx.x * 8 + wave; row < N; row += stride) {
    const int t = (int)tv[row];
    const v4f v = __builtin_nontemporal_load(
        (const v4f*)(x + (size_t)row * CDIM + lane * 4));
    const v4f m  = *(const v4f*)(s_mean + t * CDIM + lane * 4);  // ds_load_b128
    const v4f is = *(const v4f*)(s_istd + t * CDIM + lane * 4);
    const v4f r = (v - m) * is;
    __builtin_nontemporal_store(r, (v4f*)(out + (size_t)row * CDIM + lane * 4));
  }
}

// ---------------------------------------------------------------------------
extern "C" void kernel_launch(void* const* d_in, const int* in_sizes, int n_in,
                              void* d_out, int out_size, void* d_ws, size_t ws_size,
                              hipStream_t stream) {
  const float*     x  = (const float*)d_in[0];
  const long long* tv = (const long long*)d_in[1];   // int64 type ids
  const int N = in_sizes[1];                         // rows = len(type_vec)

  float* ws      = (float*)d_ws;
  float* g_sum   = ws;                               //  8192
  float* g_sq    = ws + TDIM * CDIM;                 //  8192
  float* g_cnt   = ws + 2 * TDIM * CDIM;             //    64
  float* g_stats = g_cnt + TDIM;                     // 16384 = [mean|istd]

  const int nzero = 2 * TDIM * CDIM + TDIM;
  hln_zero<<<(nzero + 255) / 256, 256, 0, stream>>>(ws, nzero);

  const size_t ldsA = (size_t)(2 * TDIM * CDIM + TDIM) * sizeof(float); // 64.25KB
  hln_partial<<<1024, 256, ldsA, stream>>>(x, tv, g_sum, g_sq, g_cnt, N);

  hln_finalize<<<(TDIM * CDIM + 255) / 256, 256, 0, stream>>>(g_sum, g_sq, g_cnt,
                                                              g_stats);

  const size_t ldsC = (size_t)(2 * TDIM * CDIM) * sizeof(float);        // 64KB
  hln_normalize<<<2048, 256, ldsC, stream>>>(x, tv, g_stats, (float*)d_out, N);
}